// EQGATBackbone_88751204205258
// MI455X (gfx1250) — compile-verified
//
#include <hip/hip_runtime.h>
#include <math.h>

#define N_NODES 20000
#define N_EDGES 320000
#define B_SEG   64
#define S_I     128
#define V_I     64
#define EDGE_DIM 32
#define EDGE_IN  292      // 2*128 + 1 + 1 + 32 + 1 + 1
#define M_TILE   32       // edges per edge-MLP block (static LDS ~55KB)
#define EPSV 1e-6f
#define CDIV(a,b) (((a)+(b)-1)/(b))

typedef __attribute__((ext_vector_type(2))) float v2f;
typedef __attribute__((ext_vector_type(8))) float v8f;

// ---------------- CDNA5 fp32 WMMA: D = A(16x4) * B(4x16) + C(16x16) ----------
__device__ __forceinline__ v8f wmma4(v2f a, v2f b, v8f c) {
  // 8 args: (neg_a, A, neg_b, B, c_mod, C, reuse_a, reuse_b)
  return __builtin_amdgcn_wmma_f32_16x16x4_f32(false, a, false, b, (short)0, c,
                                               false, false);
}

__device__ __forceinline__ float silu_f(float x) {
  return x / (1.0f + __expf(-x));
}

__device__ __forceinline__ void atomicMaxFloat(float* addr, float val) {
  unsigned int* a = (unsigned int*)addr;
  unsigned int old = __hip_atomic_load(a, __ATOMIC_RELAXED, __HIP_MEMORY_SCOPE_AGENT);
  while (__uint_as_float(old) < val) {
    unsigned int assumed = old;
    old = atomicCAS(a, assumed, __float_as_uint(val));
    if (old == assumed) break;
  }
}

__device__ __forceinline__ int lowerBound(const int* arr, int n, int key) {
  int lo = 0, hi = n;
  while (lo < hi) { int mid = (lo + hi) >> 1; if (arr[mid] < key) lo = mid + 1; else hi = mid; }
  return lo;
}

// ---------------- generic WMMA GEMM: C[M,Nc] = act(A[M,K] @ W[K,Nc] + bias) ---
// grid: (M/16, Nc/16), block: 32 (one wave per 16x16 tile). M,K,Nc multiples of 16/4.
__global__ __launch_bounds__(32)
void gemm16(const float* __restrict__ A, int lda,
            const float* __restrict__ W, int ldw,
            const float* __restrict__ bias,
            float* __restrict__ C, int ldc,
            int K, int act, int accum)
{
  int lane = threadIdx.x;
  int half = lane >> 4;
  int l15  = lane & 15;
  const float* Ap = A + ((size_t)blockIdx.x * 16 + l15) * lda;
  int ncol = blockIdx.y * 16 + l15;
  v8f acc = {0.f,0.f,0.f,0.f,0.f,0.f,0.f,0.f};
  for (int k0 = 0; k0 < K; k0 += 4) {
    int ka = k0 + 2 * half;
    v2f a, b;
    a.x = Ap[ka];
    a.y = Ap[ka + 1];
    b.x = W[(size_t)ka * ldw + ncol];
    b.y = W[(size_t)(ka + 1) * ldw + ncol];
    acc = wmma4(a, b, acc);
  }
  float bv = bias ? bias[ncol] : 0.0f;
#pragma unroll
  for (int r = 0; r < 8; ++r) {
    size_t m = (size_t)blockIdx.x * 16 + r + 8 * half;
    float val = acc[r] + bv;
    if (act == 1) val = silu_f(val);
    if (accum) C[m * ldc + ncol] += val;
    else       C[m * ldc + ncol]  = val;
  }
}

// ---------------- fused edge MLP: out = silu(aij_in@We1+be1)@We2+be2 ----------
// 32 edges per block, 256 threads (8 waves). Each wave owns a 16-col weight
// slice and reuses the B fragment across the two 16-edge M-subtiles (2 accs),
// halving weight traffic from L2. Scatters aij (+CAS atomic-max) and vij.
template <int OUTC>
__global__ __launch_bounds__(256)
void edge_mlp(const int* __restrict__ srcI, const int* __restrict__ tgtI,
              const float* __restrict__ sn, const float* __restrict__ dvec,
              const float* __restrict__ avec, const float* __restrict__ efeat,
              const float* __restrict__ dn,
              const float* __restrict__ We1, const float* __restrict__ be1,
              const float* __restrict__ We2, const float* __restrict__ be2,
              float* __restrict__ aij, float* __restrict__ vij, int vld,
              float* __restrict__ mmax)
{
  __shared__ float ain[M_TILE][EDGE_IN + 5]; // 297: odd stride -> conflict-free
  __shared__ float hls[M_TILE][S_I + 5];     // 133
  __shared__ int sSrc[M_TILE], sTgt[M_TILE];

  int tid = threadIdx.x;
  size_t e0 = (size_t)blockIdx.x * M_TILE;
  if (tid < M_TILE) {
    sSrc[tid] = srcI[e0 + tid];
    sTgt[tid] = tgtI[e0 + tid];
  }
  __syncthreads();

  for (int idx = tid; idx < M_TILE * EDGE_IN; idx += 256) {
    int row = idx / EDGE_IN, col = idx - row * EDGE_IN;
    size_t e = e0 + row;
    float val;
    if      (col < S_I)            val = sn[(size_t)sTgt[row] * S_I + col];
    else if (col < 2 * S_I)        val = sn[(size_t)sSrc[row] * S_I + (col - S_I)];
    else if (col == 2 * S_I)       val = dvec[e];
    else if (col == 2 * S_I + 1)   val = avec[e];
    else if (col < 2 * S_I + 2 + EDGE_DIM) val = efeat[e * EDGE_DIM + (col - (2 * S_I + 2))];
    else if (col == 2 * S_I + 2 + EDGE_DIM) val = dn[sTgt[row]];
    else                           val = dn[sSrc[row]];
    ain[row][col] = val;
  }
  __syncthreads();

  int wave = tid >> 5, lane = tid & 31, half = lane >> 4, l15 = lane & 15;

  // h = silu(ain @ We1 + be1)   (32 x 128); 8 N-tiles, one per wave
  {
    int nb = wave * 16 + l15;
    v8f acc0 = {0.f,0.f,0.f,0.f,0.f,0.f,0.f,0.f};
    v8f acc1 = {0.f,0.f,0.f,0.f,0.f,0.f,0.f,0.f};
    for (int k0 = 0; k0 < EDGE_IN; k0 += 4) {
      int ka = k0 + 2 * half;
      v2f b, a;
      b.x = We1[(size_t)ka * S_I + nb];
      b.y = We1[(size_t)(ka + 1) * S_I + nb];
      a.x = ain[l15][ka];      a.y = ain[l15][ka + 1];      acc0 = wmma4(a, b, acc0);
      a.x = ain[16 + l15][ka]; a.y = ain[16 + l15][ka + 1]; acc1 = wmma4(a, b, acc1);
    }
    float bv = be1[nb];
#pragma unroll
    for (int r = 0; r < 8; ++r) {
      hls[r + 8 * half][nb]      = silu_f(acc0[r] + bv);
      hls[16 + r + 8 * half][nb] = silu_f(acc1[r] + bv);
    }
  }
  __syncthreads();

  // out = h @ We2 + be2   (32 x OUTC); cols <128 -> aij (+max), else vij
  for (int nt = wave; nt < OUTC / 16; nt += 8) {
    int nb = nt * 16 + l15;
    v8f acc0 = {0.f,0.f,0.f,0.f,0.f,0.f,0.f,0.f};
    v8f acc1 = {0.f,0.f,0.f,0.f,0.f,0.f,0.f,0.f};
    for (int k0 = 0; k0 < S_I; k0 += 4) {
      int ka = k0 + 2 * half;
      v2f b, a;
      b.x = We2[(size_t)ka * OUTC + nb];
      b.y = We2[(size_t)(ka + 1) * OUTC + nb];
      a.x = hls[l15][ka];      a.y = hls[l15][ka + 1];      acc0 = wmma4(a, b, acc0);
      a.x = hls[16 + l15][ka]; a.y = hls[16 + l15][ka + 1]; acc1 = wmma4(a, b, acc1);
    }
    float bv = be2[nb];
#pragma unroll
    for (int mt = 0; mt < 2; ++mt) {
#pragma unroll
      for (int r = 0; r < 8; ++r) {
        int m = mt * 16 + r + 8 * half;
        size_t e = e0 + m;
        float val = (mt == 0 ? acc0[r] : acc1[r]) + bv;
        if (nb < S_I) {
          aij[e * S_I + nb] = val;
          atomicMaxFloat(&mmax[(size_t)sTgt[m] * S_I + nb], val);
        } else {
          vij[e * vld + (nb - S_I)] = val;
        }
      }
    }
  }
}

// ---------------- segment layernorm ------------------------------------------
__global__ __launch_bounds__(256)
void ln_stats(const float* __restrict__ s, const float* __restrict__ v,
              const int* __restrict__ index, float* __restrict__ bst)
{
  __shared__ float r1[256], r2[256], r3[256];
  __shared__ int sS, sE;
  int b = blockIdx.x, tid = threadIdx.x;
  if (tid == 0) { sS = lowerBound(index, N_NODES, b); sE = lowerBound(index, N_NODES, b + 1); }
  __syncthreads();
  int start = sS, end = sE;
  float a1 = 0.f, a2 = 0.f, a3 = 0.f;
  for (int n = start + tid; n < end; n += 256) {
    const float* sp = s + (size_t)n * S_I;
    for (int f = 0; f < S_I; ++f) { float x = sp[f]; a1 += x; a2 += x * x; }
    const float* vp = v + (size_t)n * 3 * V_I;
    for (int c = 0; c < 3 * V_I; ++c) { float x = vp[c]; a3 += x * x; }
  }
  r1[tid] = a1; r2[tid] = a2; r3[tid] = a3;
  __syncthreads();
  for (int off = 128; off > 0; off >>= 1) {
    if (tid < off) { r1[tid] += r1[tid + off]; r2[tid] += r2[tid + off]; r3[tid] += r3[tid + off]; }
    __syncthreads();
  }
  if (tid == 0) {
    float cnt = (float)((end - start) > 0 ? (end - start) : 1);
    float sm  = r1[0] / (cnt * S_I);
    float m2  = r2[0] / (cnt * S_I);
    float var = fmaxf(m2 - sm * sm, EPSV);
    float gvm = fmaxf(r3[0] / (cnt * V_I), EPSV);
    bst[b * 3 + 0] = sm;
    bst[b * 3 + 1] = 1.0f / var;
    bst[b * 3 + 2] = 1.0f / gvm;
  }
}

__global__ void ln_apply_s(float* __restrict__ s, const int* __restrict__ index,
                           const float* __restrict__ bst,
                           const float* __restrict__ w, const float* __restrict__ bb)
{
  size_t idx = (size_t)blockIdx.x * blockDim.x + threadIdx.x;
  if (idx >= (size_t)N_NODES * S_I) return;
  int n = (int)(idx >> 7), f = (int)(idx & 127);
  int b = index[n];
  s[idx] = (s[idx] - bst[b * 3]) * bst[b * 3 + 1] * w[f] + bb[f];
}

__global__ void ln_apply_v(float* __restrict__ v, const int* __restrict__ index,
                           const float* __restrict__ bst)
{
  size_t idx = (size_t)blockIdx.x * blockDim.x + threadIdx.x;
  if (idx >= (size_t)N_NODES * 192) return;
  int n = (int)(idx / 192);
  v[idx] *= bst[index[n] * 3 + 2];
}

// ---------------- softmax + segment reductions over edges --------------------
__global__ void softmax_den(const int* __restrict__ tgt, float* __restrict__ aij,
                            const float* __restrict__ mmax, float* __restrict__ den)
{
  size_t idx = (size_t)blockIdx.x * blockDim.x + threadIdx.x;
  if (idx >= (size_t)N_EDGES * S_I) return;
  size_t e = idx >> 7; int f = (int)(idx & 127);
  int t = tgt[e];
  float ex = __expf(aij[idx] - mmax[(size_t)t * S_I + f]);
  aij[idx] = ex;
  atomicAdd(&den[(size_t)t * S_I + f], ex);
}

__global__ void attn_ms(const int* __restrict__ src, const int* __restrict__ tgt,
                        const float* __restrict__ aij, const float* __restrict__ den,
                        const float* __restrict__ sb, float* __restrict__ ms)
{
  size_t idx = (size_t)blockIdx.x * blockDim.x + threadIdx.x;
  if (idx >= (size_t)N_EDGES * S_I) return;
  size_t e = idx >> 7; int f = (int)(idx & 127);
  int t = tgt[e], s = src[e];
  float att = aij[idx] / den[(size_t)t * S_I + f];
  atomicAdd(&ms[(size_t)t * S_I + f], att * sb[(size_t)s * S_I + f]);
}

template <int HASV>
__global__ void edge_mv(const int* __restrict__ src, const int* __restrict__ tgt,
                        const float* __restrict__ r3, const float* __restrict__ vij, int vld,
                        const float* __restrict__ vb, float* __restrict__ mv)
{
  size_t idx = (size_t)blockIdx.x * blockDim.x + threadIdx.x;
  if (idx >= (size_t)N_EDGES * 192) return;
  size_t e = idx / 192; int c = (int)(idx - e * 192);
  int k = c >> 6, f = c & 63;
  int t = tgt[e], s = src[e];
  float val = r3[e * 3 + k] * vij[(size_t)e * vld + f];
  if (HASV) val += vij[(size_t)e * vld + 64 + f] * vb[((size_t)s * 3 + k) * V_I + f];
  atomicAdd(&mv[((size_t)t * 3 + k) * V_I + f], val);
}

__global__ void add_ms(float* __restrict__ s, const float* __restrict__ ms) {
  size_t idx = (size_t)blockIdx.x * blockDim.x + threadIdx.x;
  if (idx < (size_t)N_NODES * S_I) s[idx] += ms[idx];
}

__global__ void add_mv(float* __restrict__ v, const float* __restrict__ mv,
                       const int* __restrict__ cnt) {
  size_t idx = (size_t)blockIdx.x * blockDim.x + threadIdx.x;
  if (idx >= (size_t)N_NODES * 192) return;
  int n = (int)(idx / 192);
  int c = cnt[n]; if (c < 1) c = 1;
  v[idx] += mv[idx] / (float)c;
}

// ---------------- gated equiv pieces -----------------------------------------
__global__ void build_scin(const float* __restrict__ s, const float* __restrict__ vv,
                           float* __restrict__ scin) {
  size_t idx = (size_t)blockIdx.x * blockDim.x + threadIdx.x;
  if (idx >= (size_t)N_NODES * 192) return;
  size_t n = idx / 192; int c = (int)(idx - n * 192);
  if (c < S_I) {
    scin[idx] = s[n * S_I + c];
  } else {
    int f = c - S_I;
    float acc = 0.f;
#pragma unroll
    for (int k = 0; k < 3; ++k) { float x = vv[(n * 3 + k) * 128 + f]; acc += x * x; }
    scin[idx] = fmaxf(acc, EPSV);
  }
}

__global__ void add_sout(float* __restrict__ s, const float* __restrict__ sc2) {
  size_t idx = (size_t)blockIdx.x * blockDim.x + threadIdx.x;
  if (idx >= (size_t)N_NODES * S_I) return;
  size_t n = idx >> 7; int f = (int)(idx & 127);
  s[idx] += sc2[n * 192 + 64 + f];
}

template <int ADDV>
__global__ void gate_mul(const float* __restrict__ sc2, const float* __restrict__ vv,
                         float* __restrict__ out) {
  size_t idx = (size_t)blockIdx.x * blockDim.x + threadIdx.x;
  if (idx >= (size_t)N_NODES * 192) return;
  size_t n = idx / 192; int c = (int)(idx - n * 192);
  int k = c >> 6, f = c & 63;
  float val = sc2[n * 192 + f] * vv[(n * 3 + k) * 128 + 64 + f];
  if (ADDV) out[idx] += val; else out[idx] = val;
}

// ---------------- init & per-layer zero --------------------------------------
__global__ void init_copy(const float* __restrict__ s, const float* __restrict__ v,
                          const float* __restrict__ p, float* __restrict__ out) {
  size_t idx = (size_t)blockIdx.x * blockDim.x + threadIdx.x;
  const size_t ns = (size_t)N_NODES * S_I, nv = (size_t)N_NODES * 192, np = (size_t)N_NODES * 3;
  if (idx < ns) out[idx] = s[idx];
  else if (idx < ns + nv) out[idx] = v[idx - ns];
  else if (idx < ns + nv + np) out[idx] = p[idx - ns - nv];
}

__global__ void init_dn(const float* __restrict__ p, float* __restrict__ dn,
                        int* __restrict__ cnt) {
  int n = blockIdx.x * blockDim.x + threadIdx.x;
  if (n < N_NODES) {
    float x = p[n * 3], y = p[n * 3 + 1], z = p[n * 3 + 2];
    dn[n] = sqrtf(fmaxf(x * x + y * y + z * z, EPSV));
    cnt[n] = 0;
  }
}

__global__ void hist_cnt(const int* __restrict__ tgt, int* __restrict__ cnt) {
  int e = blockIdx.x * blockDim.x + threadIdx.x;
  if (e < N_EDGES) atomicAdd(&cnt[tgt[e]], 1);
}

__global__ void zero_layer(float* __restrict__ ms, float* __restrict__ mv,
                           float* __restrict__ den, float* __restrict__ mmax) {
  size_t idx = (size_t)blockIdx.x * blockDim.x + threadIdx.x;
  if (idx >= (size_t)N_NODES * 192) return;
  mv[idx] = 0.f;
  if (idx < (size_t)N_NODES * S_I) { ms[idx] = 0.f; den[idx] = 0.f; mmax[idx] = -3.0e38f; }
}

// =============================================================================
struct LayerP {
  const float *We1, *We2, *Ws1, *Ws2, *Ws, *Wsc, *Wv0, *Wv1, *Wvn;
  const float *be1, *be2, *bs1, *bs2, *bs, *bsc, *normb, *normw;
  int outc;
};

extern "C" void kernel_launch(void* const* d_in, const int* in_sizes, int n_in,
                              void* d_out, int out_size, void* d_ws, size_t ws_size,
                              hipStream_t stream)
{
  const float* s_in = (const float*)d_in[0];
  const float* v_in = (const float*)d_in[1];
  const float* p_in = (const float*)d_in[2];
  const int*   eidx = (const int*)d_in[3];
  const float* dV   = (const float*)d_in[4];
  const float* aV   = (const float*)d_in[5];
  const float* rV   = (const float*)d_in[6];
  const float* eF   = (const float*)d_in[7];
  const int*   index= (const int*)d_in[8];

  const int* srcI = eidx;
  const int* tgtI = eidx + N_EDGES;

  // params: jax tree-flatten order (list order, dict keys sorted ASCII)
  LayerP LP[5];
  int pi = 9;
  for (int i = 0; i < 5; ++i) {
    LayerP& L = LP[i];
    L = LayerP{};
    auto nf = [&]() { return (const float*)d_in[pi++]; };
    L.We1 = nf(); L.We2 = nf();
    if (i < 4) { L.Ws1 = nf(); L.Ws2 = nf(); } else { L.Ws = nf(); }
    L.Wsc = nf(); L.Wv0 = nf();
    if (i < 4) L.Wv1 = nf();
    if (i > 0) L.Wvn = nf();
    L.be1 = nf(); L.be2 = nf();
    if (i < 4) { L.bs1 = nf(); L.bs2 = nf(); } else { L.bs = nf(); }
    L.bsc = nf(); L.normb = nf(); L.normw = nf();
    L.outc = (i == 0) ? 192 : 256;
  }

  // workspace carve
  float* W = (float*)d_ws;
  size_t off = 0;
  auto take = [&](size_t n) { float* p = W + off; off += n; return p; };
  float* ms   = take((size_t)N_NODES * S_I);
  float* mv   = take((size_t)N_NODES * 192);
  float* sb   = take((size_t)N_NODES * S_I);
  float* vb   = take((size_t)N_NODES * 192);
  float* dn   = take(N_NODES);
  float* mmax = take((size_t)N_NODES * S_I);
  float* den  = take((size_t)N_NODES * S_I);
  float* vv   = take((size_t)N_NODES * 3 * 128);
  float* scin = take((size_t)N_NODES * 192);
  float* t1   = take((size_t)N_NODES * S_I);
  float* sc2  = take((size_t)N_NODES * 192);
  float* vtmp = take((size_t)N_NODES * 192);
  float* bst  = take(64 * 3);
  int*   cnt  = (int*)take(N_NODES);
  float* aij  = take((size_t)N_EDGES * S_I);
  float* vij  = take((size_t)N_EDGES * 128);

  float* sS = (float*)d_out;
  float* vS = sS + (size_t)N_NODES * S_I;

  const size_t NS  = (size_t)N_NODES * S_I;
  const size_t NV  = (size_t)N_NODES * 192;
  const size_t ES  = (size_t)N_EDGES * S_I;
  const size_t EV  = (size_t)N_EDGES * 192;

  // ---- init: copy s,v,p into output stream; dn; cnt histogram
  {
    size_t tot = NS + NV + (size_t)N_NODES * 3;
    init_copy<<<CDIV(tot, 256), 256, 0, stream>>>(s_in, v_in, p_in, (float*)d_out);
    init_dn<<<CDIV(N_NODES, 256), 256, 0, stream>>>(p_in, dn, cnt);
    hist_cnt<<<CDIV(N_EDGES, 256), 256, 0, stream>>>(tgtI, cnt);
  }

  for (int i = 0; i < 5; ++i) {
    const LayerP& L = LP[i];
    bool hasv = (i > 0), mlp = (i < 4);

    // layernorm (in-place on stream)
    ln_stats<<<B_SEG, 256, 0, stream>>>(sS, vS, index, bst);
    ln_apply_s<<<CDIV(NS, 256), 256, 0, stream>>>(sS, index, bst, L.normw, L.normb);
    ln_apply_v<<<CDIV(NV, 256), 256, 0, stream>>>(vS, index, bst);

    // sb = s @ Wsc + bsc ; vb = v @ Wvn
    gemm16<<<dim3(N_NODES / 16, S_I / 16), 32, 0, stream>>>(sS, S_I, L.Wsc, S_I, L.bsc, sb, S_I, S_I, 0, 0);
    if (hasv)
      gemm16<<<dim3(3 * N_NODES / 16, V_I / 16), 32, 0, stream>>>(vS, V_I, L.Wvn, V_I, nullptr, vb, V_I, V_I, 0, 0);

    zero_layer<<<CDIV(NV, 256), 256, 0, stream>>>(ms, mv, den, mmax);

    // fused edge MLP (WMMA) + aij scatter + running max
    if (i == 0)
      edge_mlp<192><<<N_EDGES / M_TILE, 256, 0, stream>>>(srcI, tgtI, sS, dV, aV, eF, dn,
          L.We1, L.be1, L.We2, L.be2, aij, vij, 64, mmax);
    else
      edge_mlp<256><<<N_EDGES / M_TILE, 256, 0, stream>>>(srcI, tgtI, sS, dV, aV, eF, dn,
          L.We1, L.be1, L.We2, L.be2, aij, vij, 128, mmax);

    // segment softmax + segment sums
    softmax_den<<<CDIV(ES, 256), 256, 0, stream>>>(tgtI, aij, mmax, den);
    attn_ms<<<CDIV(ES, 256), 256, 0, stream>>>(srcI, tgtI, aij, den, sb, ms);
    if (hasv) edge_mv<1><<<CDIV(EV, 256), 256, 0, stream>>>(srcI, tgtI, rV, vij, 128, vb, mv);
    else      edge_mv<0><<<CDIV(EV, 256), 256, 0, stream>>>(srcI, tgtI, rV, vij, 64, vb, mv);

    add_ms<<<CDIV(NS, 256), 256, 0, stream>>>(sS, ms);
    add_mv<<<CDIV(NV, 256), 256, 0, stream>>>(vS, mv, cnt);

    // gated equiv
    gemm16<<<dim3(3 * N_NODES / 16, 128 / 16), 32, 0, stream>>>(vS, V_I, L.Wv0, 128, nullptr, vv, 128, V_I, 0, 0);
    build_scin<<<CDIV(NV, 256), 256, 0, stream>>>(sS, vv, scin);
    if (mlp) {
      gemm16<<<dim3(N_NODES / 16, S_I / 16), 32, 0, stream>>>(scin, 192, L.Ws1, S_I, L.bs1, t1, S_I, 192, 1, 0);
      gemm16<<<dim3(N_NODES / 16, 192 / 16), 32, 0, stream>>>(t1, S_I, L.Ws2, 192, L.bs2, sc2, 192, S_I, 0, 0);
    } else {
      gemm16<<<dim3(N_NODES / 16, 192 / 16), 32, 0, stream>>>(scin, 192, L.Ws, 192, L.bs, sc2, 192, 192, 0, 0);
    }
    add_sout<<<CDIV(NS, 256), 256, 0, stream>>>(sS, sc2);
    if (mlp) {
      gate_mul<0><<<CDIV(NV, 256), 256, 0, stream>>>(sc2, vv, vtmp);
      gemm16<<<dim3(3 * N_NODES / 16, V_I / 16), 32, 0, stream>>>(vtmp, V_I, L.Wv1, V_I, nullptr, vS, V_I, V_I, 0, 1);
    } else {
      gate_mul<1><<<CDIV(NV, 256), 256, 0, stream>>>(sc2, vv, vS);
    }
  }
}